// MambaSSM_49701361549628
// MI455X (gfx1250) — compile-verified
//
#include <hip/hip_runtime.h>

typedef __attribute__((ext_vector_type(2))) float v2f;
typedef __attribute__((ext_vector_type(8))) float v8f;
typedef __attribute__((ext_vector_type(4))) unsigned int u32x4;
typedef __attribute__((ext_vector_type(8))) int i32x8;
typedef __attribute__((ext_vector_type(4))) int i32x4;

#if defined(__gfx1250__) && __has_builtin(__builtin_amdgcn_tensor_load_to_lds) && \
    __has_builtin(__builtin_amdgcn_s_wait_tensorcnt)
#define USE_TDM 1
#else
#define USE_TDM 0
#endif

namespace {

constexpr int kB = 2;
constexpr int kT = 2048;
constexpr int kD = 1024;
constexpr int kN = 16;
constexpr int kL = 16;           // CHUNK
constexpr int kC = kT / kL;      // 128 chunks, sequential
constexpr int kWavesPerBlock = 8;
constexpr int kThreads = kWavesPerBlock * 32;

// Per-wave LDS: 2 buffers x 2 matrices x 256 f32 = 4 KB. 8 waves -> 32 KB/block.
constexpr int kLdsPerWaveF32 = 1024;
constexpr uint32_t kLdsPerWaveBytes = kLdsPerWaveF32 * 4;   // 4096
constexpr uint32_t kLdsBufBytes = 2048;                     // one buffer (b+c tiles)
constexpr uint32_t kLdsMatBytes = 1024;                     // one 16x16 f32 tile

#if USE_TDM
// Issue one TDM load of a 16x16 f32 tile (rows `t`, cols `n`, row stride D*N
// elements = 64KB) from global memory into LDS at byte offset `lds_byte`.
// D# fields per CDNA5 ISA 8.3/8.4. All inputs are wave-uniform -> SGPR groups.
__device__ __forceinline__ void tdm_load_tile16x16(const float* src, uint32_t lds_byte) {
  const unsigned long long ga = (unsigned long long)(uintptr_t)src;
  u32x4 g0;
  g0[0] = 1u;                                            // count=1 (valid), user mode
  g0[1] = lds_byte;                                      // lds_addr
  g0[2] = (uint32_t)ga;                                  // global_addr[31:0]
  g0[3] = (uint32_t)((ga >> 32) & 0x01FFFFFFu) | (2u << 30); // global_addr[56:32] | type=2
  i32x8 g1;
  g1[0] = 0x00020000;                                    // wg_mask=0, data_size=2 (4B)
  g1[1] = (int)(16u << 16);                              // tensor_dim0[15:0] = 16
  g1[2] = (int)(16u << 16);                              // tensor_dim1[15:0] = 16
  g1[3] = (int)(16u << 16);                              // tile_dim0 = 16
  g1[4] = 16;                                            // tile_dim1 = 16, tile_dim2 = 0
  g1[5] = kD * kN;                                       // tensor_dim0_stride = 16384 elems
  g1[6] = 0;
  g1[7] = 0;                                             // tensor_dim1_stride = 0 (2D)
  const i32x4 z4 = {0, 0, 0, 0};                         // groups 2/3 unused (<=2D tensor)
  const i32x8 z8 = {0, 0, 0, 0, 0, 0, 0, 0};             // extra group (unused), zero
  __builtin_amdgcn_tensor_load_to_lds(g0, g1, z4, z4, z8, 0);
}
#endif

// 16x16x16 f32 matmul with a 0/1 triangular matrix, built from 4 K=4 WMMA slices.
//   ANTI == false : out[m,n] = sum_{s<=m}    z[s,n]   (inclusive cumsum along L)
//   ANTI == true  : out[m,n] = sum_{s<=15-m} z[s,n]   (flipped cumsum along L)
// Tile register layout (C/D layout): lane = 16*hi + n holds z at (l = v + 8*hi, n).
template <bool ANTI>
__device__ __forceinline__ v8f tri_cumsum16(const float z[8], int hi, int m) {
  float zs[8];
#pragma unroll
  for (int v = 0; v < 8; ++v) zs[v] = __shfl_xor(z[v], 16, 32);

  v8f acc = {0.f, 0.f, 0.f, 0.f, 0.f, 0.f, 0.f, 0.f};
  auto w = [m](int s) -> float {
    return ANTI ? ((s + m <= 15) ? 1.f : 0.f) : ((s <= m) ? 1.f : 0.f);
  };

  v2f a, bb;
  int s0;

  s0 = hi ? 2 : 0;                       // K rows 0..3
  a.x = w(s0); a.y = w(s0 + 1);
  bb.x = hi ? zs[2] : z[0];
  bb.y = hi ? zs[3] : z[1];
  acc = __builtin_amdgcn_wmma_f32_16x16x4_f32(false, a, false, bb, (short)0, acc, false, false);

  s0 = hi ? 6 : 4;                       // K rows 4..7
  a.x = w(s0); a.y = w(s0 + 1);
  bb.x = hi ? zs[6] : z[4];
  bb.y = hi ? zs[7] : z[5];
  acc = __builtin_amdgcn_wmma_f32_16x16x4_f32(false, a, false, bb, (short)0, acc, false, false);

  s0 = hi ? 10 : 8;                      // K rows 8..11
  a.x = w(s0); a.y = w(s0 + 1);
  bb.x = hi ? z[2] : zs[0];
  bb.y = hi ? z[3] : zs[1];
  acc = __builtin_amdgcn_wmma_f32_16x16x4_f32(false, a, false, bb, (short)0, acc, false, false);

  s0 = hi ? 14 : 12;                     // K rows 12..15
  a.x = w(s0); a.y = w(s0 + 1);
  bb.x = hi ? z[6] : zs[4];
  bb.y = hi ? z[7] : zs[5];
  acc = __builtin_amdgcn_wmma_f32_16x16x4_f32(false, a, false, bb, (short)0, acc, false, false);

  return acc;
}

}  // namespace

__global__ __launch_bounds__(kThreads) void mamba_chunk_scan(
    const float* __restrict__ u, const float* __restrict__ delta,
    const float* __restrict__ bmat, const float* __restrict__ cmat,
    const float* __restrict__ alog, float* __restrict__ out) {
  __shared__ float lds_tiles[kWavesPerBlock * kLdsPerWaveF32];

  const int lane = threadIdx.x & 31;
  const int hi   = lane >> 4;   // which half-wave: owns L rows [0..7] or [8..15]
  const int n    = lane & 15;   // state index (column)
  const int wslot = __builtin_amdgcn_readfirstlane((int)(threadIdx.x >> 5));
  const int wid  = blockIdx.x * kWavesPerBlock + wslot;
  const int b    = wid / kD;
  const int d    = wid - b * kD;

  const float an  = -__expf(alog[n]);   // A[n] = -exp(A_log[n])
  const float anp = an + 1e-12f;

  float h = 0.f;  // carried state h[b,d,n] (replicated in both half-waves)

#if USE_TDM
  // Escape the LDS array so the compiler treats the TDM DMA writes as real.
  asm volatile("" : : "v"(&lds_tiles[0]) : "memory");
  const uint32_t lds_wave = (uint32_t)wslot * kLdsPerWaveBytes;
  auto issue_chunk = [&](int c, int buf) {
    const size_t e0 = (((size_t)(b * kT + c * kL)) * kD + d) * kN;
    tdm_load_tile16x16(bmat + e0, lds_wave + (uint32_t)buf * kLdsBufBytes);
    tdm_load_tile16x16(cmat + e0, lds_wave + (uint32_t)buf * kLdsBufBytes + kLdsMatBytes);
  };
  issue_chunk(0, 0);  // preload chunk 0 into buffer 0
#endif

  for (int c = 0; c < kC; ++c) {
    const int t0 = c * kL;

#if USE_TDM
    if (c + 1 < kC) issue_chunk(c + 1, (c + 1) & 1);   // overlap DMA with compute
#endif

    size_t row[8];
    float dl[8], uu[8], bt[8], ct[8];
#pragma unroll
    for (int v = 0; v < 8; ++v) {
      const int l = v + 8 * hi;
      row[v] = ((size_t)(b * kT + t0 + l)) * kD + d;
      dl[v] = delta[row[v]];            // half-wave uniform -> broadcast
      uu[v] = u[row[v]];
    }
    // Prefetch next chunk's u/delta rows -> global_prefetch_b8
    if (c + 1 < kC) {
#pragma unroll
      for (int v = 0; v < 8; ++v) {
        const size_t nrow = row[v] + (size_t)kL * kD;
        __builtin_prefetch(delta + nrow, 0, 0);
        __builtin_prefetch(u + nrow, 0, 0);
      }
    }

#if USE_TDM
    // TDM completes in-order: waiting <=2 guarantees chunk c's two tiles landed,
    // while chunk c+1's two loads stay in flight.
    if (c + 1 < kC) __builtin_amdgcn_s_wait_tensorcnt(2);
    else            __builtin_amdgcn_s_wait_tensorcnt(0);
    asm volatile("" ::: "memory");
    {
      const int tb = wslot * kLdsPerWaveF32 + (c & 1) * 512;
#pragma unroll
      for (int v = 0; v < 8; ++v) {
        const int l16 = (v + 8 * hi) * 16 + n;
        bt[v] = lds_tiles[tb + l16];          // b tile
        ct[v] = lds_tiles[tb + 256 + l16];    // c tile
      }
    }
#else
#pragma unroll
    for (int v = 0; v < 8; ++v) {
      bt[v] = bmat[row[v] * kN + n];
      ct[v] = cmat[row[v] * kN + n];
    }
#endif

    float dA[8], zz[8];
#pragma unroll
    for (int v = 0; v < 8; ++v) {
      float x = dl[v] * an;
      x = fminf(fmaxf(x, -10.f), 10.f);           // clip(delta*A, -10, 10)
      dA[v] = x;
      const float eA   = __expf(x);
      const float frac = (fabsf(x) < 1e-4f) ? dl[v] : (eA - 1.f) / anp;
      const float Bu   = frac * bt[v] * uu[v];
      const bool  last = (v == 7) && (hi == 1);   // l == 15 -> fw = 1
      zz[v] = Bu * (last ? 1.f : eA);             // Bu * fw
    }

    // cumprod(eA) == exp(cumsum(dA));  Bu_term == anti-triangular sums of zz
    const v8f S  = tri_cumsum16<false>(dA, hi, n);
    const v8f BT = tri_cumsum16<true >(zz, hi, n);

    float h15 = 0.f;
#pragma unroll
    for (int v = 0; v < 8; ++v) {
      const float hn = h * __expf(S[v]) + BT[v];  // h_expand + Bu_term
      if (v == 7) h15 = hn;                       // l==15 value lives in hi==1 lanes
      float y = hn * ct[v];                       // reduce over n (16 lanes / half)
      y += __shfl_xor(y, 1, 32);
      y += __shfl_xor(y, 2, 32);
      y += __shfl_xor(y, 4, 32);
      y += __shfl_xor(y, 8, 32);
      if (n == 0) out[row[v]] = y;
    }
    // carry h_new at l==15: broadcast from lane n+16 to both halves
    h = __shfl(h15, n + 16, 32);
  }
}

extern "C" void kernel_launch(void* const* d_in, const int* in_sizes, int n_in,
                              void* d_out, int out_size, void* d_ws, size_t ws_size,
                              hipStream_t stream) {
  (void)in_sizes; (void)n_in; (void)d_ws; (void)ws_size; (void)out_size;
  const float* u     = (const float*)d_in[0];
  const float* delta = (const float*)d_in[1];
  const float* bmat  = (const float*)d_in[2];
  const float* cmat  = (const float*)d_in[3];
  const float* alog  = (const float*)d_in[4];
  float* out = (float*)d_out;

  const int waves = kB * kD;                 // one wave per (b, d)
  dim3 grid(waves / kWavesPerBlock);         // 256 blocks x 256 threads
  mamba_chunk_scan<<<grid, kThreads, 0, stream>>>(u, delta, bmat, cmat, alog, out);
}